// NPU_88029649698990
// MI455X (gfx1250) — compile-verified
//
#include <hip/hip_runtime.h>

typedef float v2f __attribute__((ext_vector_type(2)));
typedef float v8f __attribute__((ext_vector_type(8)));

#define IN_DIM    128
#define OUT_DIM   128
#define OUT_TILE  16
#define ROW_TILE  64          // rows per block (4 waves x 16 rows)
#define SX_STRIDE 132         // 128 + 4 pad: keeps 16B alignment, spreads banks 4*lane

__global__ __launch_bounds__(128) void NPU_phase_kernel(
    const float* __restrict__ x,  const float* __restrict__ Wr,
    const float* __restrict__ Wi, const float* __restrict__ G,
    float* __restrict__ out, int n)
{
    __shared__ float sX [ROW_TILE * SX_STRIDE];   // 33792 B
    __shared__ float sG [IN_DIM * OUT_TILE];      //  8192 B
    __shared__ float sWr[IN_DIM * OUT_TILE];      //  8192 B
    __shared__ float sWi[IN_DIM * OUT_TILE];      //  8192 B

    const int n0 = blockIdx.x * ROW_TILE;
    const int o0 = blockIdx.y * OUT_TILE;
    if (n0 >= n) return;

    // ---- stage x tile (64 x 128 f32), float4 coalesced ----
    const float4* xsrc = (const float4*)(x + (size_t)n0 * IN_DIM);
    for (int v = threadIdx.x; v < ROW_TILE * (IN_DIM / 4); v += blockDim.x) {
        int row = v >> 5;               // 32 float4 per row
        int c4  = v & 31;
        float4 val = xsrc[(size_t)row * (IN_DIM / 4) + c4];
        *(float4*)&sX[row * SX_STRIDE + c4 * 4] = val;
    }
    // ---- stage weight tiles (128 x 16 each) ----
    for (int idx = threadIdx.x; idx < IN_DIM * OUT_TILE; idx += blockDim.x) {
        int i  = idx >> 4;
        int oo = idx & 15;
        float g = G[i * OUT_DIM + o0 + oo];
        g = fminf(fmaxf(g, 0.0f), 1.0f);          // clip(G,0,1)
        sG [idx] = g;
        sWr[idx] = Wr[i * OUT_DIM + o0 + oo];
        sWi[idx] = Wi[i * OUT_DIM + o0 + oo];
    }
    __syncthreads();

    const int lane    = threadIdx.x & 31;
    const int wave    = threadIdx.x >> 5;     // 0..3, owns rows [wave*16, wave*16+16)
    const int half    = lane >> 4;            // WMMA C layout: lanes>=16 hold M+8
    const int ocol    = lane & 15;            // WMMA C layout: N = lane&15
    const int rowbase = wave * 16 + half * 8;

    // ---------------- main VALU log-domain accumulation ----------------
    // sumR[r], sumI[r] correspond to C/D element r of the 16x16 f32 WMMA layout.
    float sumR[8], sumI[8];
    #pragma unroll
    for (int r = 0; r < 8; ++r) { sumR[r] = 0.0f; sumI[r] = 0.0f; }

    #pragma unroll 2
    for (int i = 0; i < IN_DIM; ++i) {
        float g   = sG [i * OUT_TILE + ocol];
        float wr  = sWr[i * OUT_TILE + ocol];
        float wi  = sWi[i * OUT_TILE + ocol];
        // (|x|+eps)*g + (1-g) == fma(|x|, g, fma(eps, g, 1-g))
        float omg = fmaf(1e-36f, g, 1.0f - g);
        #pragma unroll
        for (int r = 0; r < 8; ++r) {
            float xv = sX[(rowbase + r) * SX_STRIDE + i];   // broadcast across 16 lanes
            float t  = fmaf(fabsf(xv), g, omg);
            float l  = __log2f(t);                          // native v_log_f32
            sumR[r]  = fmaf(l, wr, sumR[r]);
            sumI[r]  = fmaf(l, wi, sumI[r]);
        }
    }

    // ---------------- sign GEMMs on the matrix pipe ----------------
    // accR = S @ (Gc .* Wr), accI = S @ (Gc .* Wi), S[n,i] = (x[n,i] < 0)
    v8f accR = {};
    v8f accI = {};
    const int mrow = wave * 16 + ocol;    // A-matrix row M for this lane
    for (int k0 = 0; k0 < IN_DIM; k0 += 4) {
        // A frag (16x4 f32): lanes 0-15 hold K = k0,k0+1 ; lanes 16-31 hold K = k0+2,k0+3
        int ka = k0 + half * 2;
        float x0 = sX[mrow * SX_STRIDE + ka];
        float x1 = sX[mrow * SX_STRIDE + ka + 1];
        v2f a;
        a.x = (x0 < 0.0f) ? 1.0f : 0.0f;
        a.y = (x1 < 0.0f) ? 1.0f : 0.0f;
        // B frag (4x16 f32): lane gives N=ocol; same K split across halves
        float g0 = sG[ka * OUT_TILE + ocol];
        float g1 = sG[(ka + 1) * OUT_TILE + ocol];
        v2f bR, bI;
        bR.x = g0 * sWr[ ka      * OUT_TILE + ocol];
        bR.y = g1 * sWr[(ka + 1) * OUT_TILE + ocol];
        bI.x = g0 * sWi[ ka      * OUT_TILE + ocol];
        bI.y = g1 * sWi[(ka + 1) * OUT_TILE + ocol];
        accR = __builtin_amdgcn_wmma_f32_16x16x4_f32(false, a, false, bR,
                                                     (short)0, accR, false, false);
        accI = __builtin_amdgcn_wmma_f32_16x16x4_f32(false, a, false, bI,
                                                     (short)0, accI, false, false);
    }

    // ---------------- epilogue: fold constants, exp2/cos ----------------
    // A = ln2*sumR - pi*ki   -> exp(A) = exp2(sumR - (pi*log2e)*ki)
    // B = ln2*sumI + pi*kr
    const float LN2      = 0.69314718055994530942f;
    const float PI_F     = 3.14159265358979323846f;
    const float PI_LOG2E = 4.53236014182719380962f;   // pi / ln2

    #pragma unroll
    for (int r = 0; r < 8; ++r) {
        float Aarg = sumR[r] - PI_LOG2E * accI[r];
        float Barg = fmaf(LN2, sumI[r], PI_F * accR[r]);
        float val  = exp2f(Aarg) * __cosf(Barg);       // exp2 is native on AMD
        out[(size_t)(n0 + rowbase + r) * OUT_DIM + (o0 + ocol)] = val;
    }
}

extern "C" void kernel_launch(void* const* d_in, const int* in_sizes, int n_in,
                              void* d_out, int out_size, void* d_ws, size_t ws_size,
                              hipStream_t stream) {
    (void)n_in; (void)out_size; (void)d_ws; (void)ws_size;
    const float* x  = (const float*)d_in[0];
    const float* Wr = (const float*)d_in[1];
    const float* Wi = (const float*)d_in[2];
    const float* G  = (const float*)d_in[3];
    float* out = (float*)d_out;

    int n = in_sizes[0] / IN_DIM;                  // 8192
    dim3 grid((n + ROW_TILE - 1) / ROW_TILE,       // 128
              OUT_DIM / OUT_TILE);                 // 8
    dim3 block(128);                               // 4 waves (wave32)
    NPU_phase_kernel<<<grid, block, 0, stream>>>(x, Wr, Wi, G, out, n);
}